// Controller_48773648613557
// MI455X (gfx1250) — compile-verified
//
#include <hip/hip_runtime.h>
#include <hip/hip_bf16.h>
#include <math.h>

// Problem sizes (fixed by the reference)
#define BATCH 32
#define MROWS 32768
#define DIM   256
#define TOPK  16
#define TEMP  5.0f
#define BSCALE 32

// d_out flat layout (float32): best_fp[32] | logits[32*32768] | topk_idx[32*16] | topk_w[32*16]
#define OFF_LOGITS 32
#define OFF_IDX    (32 + BATCH * MROWS)          // 1048608
#define OFF_W      (OFF_IDX + BATCH * TOPK)      // 1049120

typedef __attribute__((ext_vector_type(2))) float v2f;
typedef __attribute__((ext_vector_type(4))) float v4f;
typedef __attribute__((ext_vector_type(8))) float v8f;

// ---------------------------------------------------------------------------
// Kernel 1: q = state @ qp_w^T + qp_b   (32x256 = A[32x256] * B[256x256])
// One wave per 16x16 output tile, V_WMMA_F32_16X16X4_F32, K stepped by 4.
// A layout: lane L holds A[M=L&15, K=k0+2*(L>>4) + {0,1}] in a v2f.
// B layout (mirrors A): lane L holds B[K=k0+2*(L>>4)+{0,1}, N=L&15].
// C/D layout: VGPR v on lane L = D[M = v + 8*(L>>4), N = L&15].
// ---------------------------------------------------------------------------
__global__ __launch_bounds__(32) void proj_q_kernel(const float* __restrict__ state,
                                                    const float* __restrict__ qp_w,
                                                    const float* __restrict__ qp_b,
                                                    float* __restrict__ q) {
    const int lane   = threadIdx.x;
    const int mt     = blockIdx.x >> 4;    // 0..1   (32 batch rows / 16)
    const int nt     = blockIdx.x & 15;    // 0..15  (256 out cols / 16)
    const int m_base = mt * 16;
    const int n_base = nt * 16;
    const int mr     = lane & 15;          // A row within tile / B col within tile
    const int kh     = (lane >> 4) * 2;    // K sub-offset: 0 (lanes 0-15) or 2 (16-31)

    v8f acc = {};
    for (int k0 = 0; k0 < DIM; k0 += 4) {
        // A[m, k] = state[m_base+mr, k0+kh .. +1]
        v2f a = *(const v2f*)(state + (m_base + mr) * DIM + k0 + kh);
        // B[k, n] = qp_w[n, k] (qp_w is [out,in] row-major -> contiguous in k)
        v2f b = *(const v2f*)(qp_w + (n_base + mr) * DIM + k0 + kh);
        acc = __builtin_amdgcn_wmma_f32_16x16x4_f32(false, a, false, b,
                                                    (short)0, acc, false, false);
    }
    const float bias  = qp_b[n_base + mr];
    const int   mrow0 = m_base + ((lane >> 4) << 3);
#pragma unroll
    for (int v = 0; v < 8; ++v)
        q[(mrow0 + v) * DIM + n_base + mr] = acc[v] + bias;
}

// ---------------------------------------------------------------------------
// Kernel 2: r = q @ kp_w  (r[b,d] = sum_e q[b,e]*kp_w[e,d]),  c[b] = q[b]·kp_b
// ---------------------------------------------------------------------------
__global__ __launch_bounds__(32) void proj_r_kernel(const float* __restrict__ q,
                                                    const float* __restrict__ kp_w,
                                                    const float* __restrict__ kp_b,
                                                    float* __restrict__ r,
                                                    float* __restrict__ cbias) {
    const int lane   = threadIdx.x;
    const int mt     = blockIdx.x >> 4;
    const int nt     = blockIdx.x & 15;
    const int m_base = mt * 16;
    const int n_base = nt * 16;
    const int mr     = lane & 15;
    const int kh     = (lane >> 4) * 2;

    v8f acc = {};
    for (int k0 = 0; k0 < DIM; k0 += 4) {
        v2f a = *(const v2f*)(q + (m_base + mr) * DIM + k0 + kh);
        // B[k, n] = kp_w[k, n] (kp_w is [e,d] row-major -> strided in k)
        v2f b;
        b.x = kp_w[(k0 + kh) * DIM + n_base + mr];
        b.y = kp_w[(k0 + kh + 1) * DIM + n_base + mr];
        acc = __builtin_amdgcn_wmma_f32_16x16x4_f32(false, a, false, b,
                                                    (short)0, acc, false, false);
    }
    const int mrow0 = m_base + ((lane >> 4) << 3);
#pragma unroll
    for (int v = 0; v < 8; ++v)
        r[(mrow0 + v) * DIM + n_base + mr] = acc[v];

    // One tile-column of blocks also computes the scalar bias term c[b] = q[b]·kp_b
    if (nt == 0 && lane < 16) {
        const float* qr = q + (m_base + lane) * DIM;
        float s = 0.f;
        for (int e = 0; e < DIM; ++e) s = fmaf(qr[e], kp_b[e], s);
        cbias[m_base + lane] = s;
    }
}

// ---------------------------------------------------------------------------
// Kernel 3 (the bandwidth-bound one): s[b,m] = (pm[b,m]·r[b] + c[b]) / 16
// 4096 blocks of 256 threads; each wave streams 32 rows with coalesced b128
// non-temporal loads (pm = 1.07 GB, don't pollute L2), r held in registers,
// wave32 shuffle reduction.
// ---------------------------------------------------------------------------
__global__ __launch_bounds__(256) void score_kernel(const float* __restrict__ pm,
                                                    const float* __restrict__ r,
                                                    const float* __restrict__ cbias,
                                                    float* __restrict__ logits) {
    const int b     = blockIdx.x >> 7;     // 128 chunks per batch
    const int chunk = blockIdx.x & 127;
    const int lane  = threadIdx.x & 31;
    const int wave  = threadIdx.x >> 5;

    const v4f r0 = *(const v4f*)(r + b * DIM + lane * 4);
    const v4f r1 = *(const v4f*)(r + b * DIM + lane * 4 + 128);
    const float cb = cbias[b];

    const float* pmb = pm + (size_t)b * MROWS * DIM;
    const int row_base = chunk * 256 + wave * 32;

    for (int t = 0; t < 32; ++t) {
        const int row = row_base + t;
        const float* p = pmb + (size_t)row * DIM + lane * 4;
        v4f x0 = __builtin_nontemporal_load((const v4f*)p);
        v4f x1 = __builtin_nontemporal_load((const v4f*)(p + 128));
        float acc = x0.x * r0.x;
        acc = fmaf(x0.y, r0.y, acc);
        acc = fmaf(x0.z, r0.z, acc);
        acc = fmaf(x0.w, r0.w, acc);
        acc = fmaf(x1.x, r1.x, acc);
        acc = fmaf(x1.y, r1.y, acc);
        acc = fmaf(x1.z, r1.z, acc);
        acc = fmaf(x1.w, r1.w, acc);
#pragma unroll
        for (int off = 16; off > 0; off >>= 1)
            acc += __shfl_xor(acc, off, 32);
        if (lane == 0)
            logits[(size_t)b * MROWS + row] = (acc + cb) * 0.0625f;  // 1/sqrt(256)
    }
}

// ---------------------------------------------------------------------------
// Kernel 4: per-batch top-16 (sorted desc, ties -> smaller index, like
// lax.top_k), softmax(vals/5), best_fp = idx[0]*32.  One block per batch.
// Per-thread register top-16 over 128 strided elems, then LDS tree-merge.
// ---------------------------------------------------------------------------
__global__ __launch_bounds__(256) void topk_kernel(const float* __restrict__ logits,
                                                   float* __restrict__ best_fp,
                                                   float* __restrict__ out_idx,
                                                   float* __restrict__ out_w) {
    __shared__ float sval[256 * TOPK];
    __shared__ int   sidx[256 * TOPK];

    const int b   = blockIdx.x;
    const int tid = threadIdx.x;
    const float* lg = logits + (size_t)b * MROWS;

    float v[TOPK];
    int   id[TOPK];
#pragma unroll
    for (int i = 0; i < TOPK; ++i) { v[i] = -INFINITY; id[i] = 0x7fffffff; }

    for (int m = tid; m < MROWS; m += 256) {
        const float x = lg[m];
        if (x > v[TOPK - 1]) {               // qualifies: insert & bubble up
            v[TOPK - 1] = x; id[TOPK - 1] = m;
#pragma unroll
            for (int p = TOPK - 1; p > 0; --p) {
                if (v[p] > v[p - 1]) {       // strict: earlier (smaller) idx wins ties
                    float tv = v[p]; v[p] = v[p - 1]; v[p - 1] = tv;
                    int   ti = id[p]; id[p] = id[p - 1]; id[p - 1] = ti;
                }
            }
        }
    }
#pragma unroll
    for (int i = 0; i < TOPK; ++i) { sval[tid * TOPK + i] = v[i]; sidx[tid * TOPK + i] = id[i]; }
    __syncthreads();

    // Tree-merge 256 sorted lists -> 1 (8 levels); two-pointer merge keeps top 16.
    for (int stride = 128; stride >= 1; stride >>= 1) {
        if (tid < stride) {
            const int aB = tid * TOPK, bB = (tid + stride) * TOPK;
            float mv[TOPK]; int mi[TOPK];
            int i = 0, j = 0;
#pragma unroll
            for (int o = 0; o < TOPK; ++o) {
                const float va = sval[aB + i]; const int ia = sidx[aB + i];
                const float vb = sval[bB + j]; const int ib = sidx[bB + j];
                const bool takeA = (va > vb) || (va == vb && ia <= ib);
                mv[o] = takeA ? va : vb;
                mi[o] = takeA ? ia : ib;
                i += takeA ? 1 : 0;
                j += takeA ? 0 : 1;
            }
#pragma unroll
            for (int o = 0; o < TOPK; ++o) { sval[aB + o] = mv[o]; sidx[aB + o] = mi[o]; }
        }
        __syncthreads();
    }

    if (tid == 0) {
        const float vmax = sval[0];          // sorted descending
        float e[TOPK], sum = 0.f;
#pragma unroll
        for (int i = 0; i < TOPK; ++i) { e[i] = __expf((sval[i] - vmax) / TEMP); sum += e[i]; }
        const float inv = 1.0f / sum;
        best_fp[b] = (float)(sidx[0] * BSCALE);
#pragma unroll
        for (int i = 0; i < TOPK; ++i) {
            out_idx[b * TOPK + i] = (float)sidx[i];
            out_w[b * TOPK + i]   = e[i] * inv;
        }
    }
}

// ---------------------------------------------------------------------------
extern "C" void kernel_launch(void* const* d_in, const int* in_sizes, int n_in,
                              void* d_out, int out_size, void* d_ws, size_t ws_size,
                              hipStream_t stream) {
    const float* pm    = (const float*)d_in[0];
    const float* state = (const float*)d_in[1];
    const float* qp_w  = (const float*)d_in[2];
    const float* qp_b  = (const float*)d_in[3];
    const float* kp_w  = (const float*)d_in[4];
    const float* kp_b  = (const float*)d_in[5];
    float* out = (float*)d_out;

    // Workspace: q[32*256] | r[32*256] | c[32]   (~66 KB)
    float* q  = (float*)d_ws;
    float* r  = q + BATCH * DIM;
    float* cb = r + BATCH * DIM;

    proj_q_kernel<<<32, 32, 0, stream>>>(state, qp_w, qp_b, q);
    proj_r_kernel<<<32, 32, 0, stream>>>(q, kp_w, kp_b, r, cb);
    score_kernel<<<BATCH * (MROWS / 256), 256, 0, stream>>>(pm, r, cb, out + OFF_LOGITS);
    topk_kernel<<<BATCH, 256, 0, stream>>>(out + OFF_LOGITS, out, out + OFF_IDX, out + OFF_W);
}